// NCCVxm_47029891891776
// MI455X (gfx1250) — compile-verified
//
#include <hip/hip_runtime.h>

typedef __attribute__((ext_vector_type(2))) float v2f;
typedef __attribute__((ext_vector_type(8))) float v8f;

#if __has_builtin(__builtin_amdgcn_wmma_f32_16x16x4_f32)
#define HAVE_WMMA_F32X4 1
#else
#define HAVE_WMMA_F32X4 0
#endif

#if __has_builtin(__builtin_amdgcn_global_load_async_to_lds_b32) && \
    __has_builtin(__builtin_amdgcn_s_wait_asynccnt)
#define HAVE_ASYNC_LDS 1
#else
#define HAVE_ASYNC_LDS 0
#endif

typedef __attribute__((address_space(1))) int* gptr_i32;
typedef __attribute__((address_space(3))) int* lptr_i32;

// Problem constants
#define NB   2
#define DD   160
#define HH   192
#define WW   224
#define TILE 16
#define PAD  4
#define PT   24            // padded tile edge
#define PT3  (PT*PT*PT)    // 13824
#define NT   256           // threads per block
#define TD   (DD/TILE)     // 10
#define TH   (HH/TILE)     // 12
#define TW   (WW/TILE)     // 14
#define NBLK (NB*TD*TH*TW) // 3360
#define WINSZ 729.0f
#define EPSV  1e-5f

__global__ __launch_bounds__(NT, 1)
void ncc_fused_kernel(const float* __restrict__ gI,
                      const float* __restrict__ gJ,
                      float* __restrict__ out)
{
    __shared__ float rawI[PT3];
    __shared__ float rawJ[PT3];
    __shared__ float scratch[PT3];
    __shared__ float interior[5][TILE*TILE*TILE];
    __shared__ float redsum;

    const int tid  = threadIdx.x;
    const int lane = tid & 31;
    const int wave = tid >> 5;

    // ---- tile decode ----
    int b   = blockIdx.x;
    int n   = b / (TD*TH*TW);
    int rem = b % (TD*TH*TW);
    int td  = rem / (TH*TW);
    int r2  = rem % (TH*TW);
    int th  = r2 / TW;
    int tw  = r2 % TW;
    int d0 = td*TILE, h0 = th*TILE, w0 = tw*TILE;
    long base = (long)n * (long)(DD*HH*WW);

    // ---- stage padded raw tiles (zero-pad halo, matches conv zero padding) ----
#if HAVE_ASYNC_LDS
    // Async global->LDS DMA for in-volume elements (ASYNCcnt domain);
    // plain DS stores zero-fill the disjoint halo set (DScnt domain).
    for (int idx = tid; idx < PT3; idx += NT) {
        int pd = idx / (PT*PT);
        int rr = idx % (PT*PT);
        int ph = rr / PT;
        int pw = rr % PT;
        int gd = d0 + pd - PAD, gh = h0 + ph - PAD, gw = w0 + pw - PAD;
        bool in = ((unsigned)gd < (unsigned)DD) &&
                  ((unsigned)gh < (unsigned)HH) &&
                  ((unsigned)gw < (unsigned)WW);
        long off = base + ((long)gd*HH + gh)*WW + gw;
        if (in) {
            __builtin_amdgcn_global_load_async_to_lds_b32(
                (gptr_i32)(gI + off), (lptr_i32)(rawI + idx), 0, 0);
            __builtin_amdgcn_global_load_async_to_lds_b32(
                (gptr_i32)(gJ + off), (lptr_i32)(rawJ + idx), 0, 0);
        } else {
            rawI[idx] = 0.0f;
            rawJ[idx] = 0.0f;
        }
    }
    __builtin_amdgcn_s_wait_asynccnt(0);
    __syncthreads();
#else
    for (int idx = tid; idx < PT3; idx += NT) {
        int pd = idx / (PT*PT);
        int rr = idx % (PT*PT);
        int ph = rr / PT;
        int pw = rr % PT;
        int gd = d0 + pd - PAD, gh = h0 + ph - PAD, gw = w0 + pw - PAD;
        bool in = ((unsigned)gd < (unsigned)DD) &&
                  ((unsigned)gh < (unsigned)HH) &&
                  ((unsigned)gw < (unsigned)WW);
        long off = base + ((long)gd*HH + gh)*WW + gw;
        rawI[idx] = in ? gI[off] : 0.0f;
        rawJ[idx] = in ? gJ[off] : 0.0f;
    }
    __syncthreads();
#endif

#if HAVE_WMMA_F32X4
    // ---- constant band matrix B (24x16 split into 6 K-chunks of 4), f32 WMMA layout:
    // B 4x16: lanes 0-15 -> N=lane, K={0,1} in v0/v1 ; lanes 16-31 -> K={2,3}
    v2f bmat[6];
    {
        int nn    = lane & 15;
        int khalf = (lane < 16) ? 0 : 2;
#pragma unroll
        for (int t = 0; t < 6; ++t) {
            int kg = 4*t + khalf;
            bmat[t].x = (kg   >= nn && kg   <= nn + 8) ? 1.0f : 0.0f;
            bmat[t].y = (kg+1 >= nn && kg+1 <= nn + 8) ? 1.0f : 0.0f;
        }
    }
#endif

    // field order: I^2, J^2, I*J (need raws intact, use scratch), then I, J in place
    for (int step = 0; step < 5; ++step) {
        float* buf;
        int fi;
        if      (step == 0) { buf = scratch; fi = 2; }
        else if (step == 1) { buf = scratch; fi = 3; }
        else if (step == 2) { buf = scratch; fi = 4; }
        else if (step == 3) { buf = rawI;    fi = 0; }
        else                { buf = rawJ;    fi = 1; }

        if (step < 3) {
            for (int idx = tid; idx < PT3; idx += NT) {
                float a = rawI[idx], c = rawJ[idx];
                scratch[idx] = (step == 0) ? a*a : (step == 1) ? c*c : a*c;
            }
            __syncthreads();
        }

        // ---------- W sweep: banded matmul on the matrix unit ----------
#if HAVE_WMMA_F32X4
        {
            // 576 rows (pd,ph), groups of 16 rows, A 16x24 * Band 24x16 = D 16x16
            int m     = lane & 15;
            int khalf = (lane < 16) ? 0 : 2;
            for (int g = wave; g < 36; g += 8) {     // uniform per wave; EXEC all ones
                v2f a[6];
#pragma unroll
                for (int t = 0; t < 6; ++t) {
                    int k0 = 4*t + khalf;            // even -> 8B aligned
                    a[t] = *(const v2f*)(buf + (g*16 + m)*PT + k0);
                }
                v8f c = {};
#pragma unroll
                for (int t = 0; t < 6; ++t)
                    c = __builtin_amdgcn_wmma_f32_16x16x4_f32(
                            false, a[t], false, bmat[t], (short)0, c, false, false);
#pragma unroll
                for (int r = 0; r < 8; ++r) {
                    int mr = r + ((lane < 16) ? 0 : 8);
                    buf[(g*16 + mr)*PT + PAD + m] = c[r];
                }
            }
        }
#else
        {
            for (int r = tid; r < PT*PT; r += NT) {  // row (pd,ph), contiguous in W
                float x[PT];
#pragma unroll
                for (int k = 0; k < PT; ++k) x[k] = buf[r*PT + k];
                float s = 0.0f;
#pragma unroll
                for (int k = 0; k < 9; ++k) s += x[k];
                buf[r*PT + PAD] = s;
#pragma unroll
                for (int i = 1; i < TILE; ++i) { s += x[i+8] - x[i-1]; buf[r*PT + PAD + i] = s; }
            }
        }
#endif
        __syncthreads();

        // ---------- H sweep: per-thread sliding sum, in place ----------
        for (int r = tid; r < PT*TILE; r += NT) {    // (pd, interior pw)
            int pd = r >> 4;
            int pw = PAD + (r & 15);
            float x[PT];
#pragma unroll
            for (int ph = 0; ph < PT; ++ph) x[ph] = buf[(pd*PT + ph)*PT + pw];
            float s = 0.0f;
#pragma unroll
            for (int k = 0; k < 9; ++k) s += x[k];
            buf[(pd*PT + PAD)*PT + pw] = s;
#pragma unroll
            for (int i = 1; i < TILE; ++i) { s += x[i+8] - x[i-1]; buf[(pd*PT + PAD + i)*PT + pw] = s; }
        }
        __syncthreads();

        // ---------- D sweep: 256 interior columns, one per thread ----------
        {
            int lh = tid >> 4, lw = tid & 15;
            int ph = PAD + lh, pw = PAD + lw;
            float x[PT];
#pragma unroll
            for (int pd = 0; pd < PT; ++pd) x[pd] = buf[(pd*PT + ph)*PT + pw];
            float s = 0.0f;
#pragma unroll
            for (int k = 0; k < 9; ++k) s += x[k];
            interior[fi][(0*TILE + lh)*TILE + lw] = s;
#pragma unroll
            for (int i = 1; i < TILE; ++i) {
                s += x[i+8] - x[i-1];
                interior[fi][(i*TILE + lh)*TILE + lw] = s;
            }
        }
        __syncthreads();
    }

    // ---------- NCC pointwise + reduction ----------
    float local = 0.0f;
    for (int idx = tid; idx < TILE*TILE*TILE; idx += NT) {
        float isum = interior[0][idx];
        float jsum = interior[1][idx];
        float i2s  = interior[2][idx];
        float j2s  = interior[3][idx];
        float ijs  = interior[4][idx];
        float ui = isum * (1.0f / WINSZ);
        float uj = jsum * (1.0f / WINSZ);
        float cross = ijs - uj*isum - ui*jsum + ui*uj*WINSZ;
        float ivar  = fmaxf(i2s - 2.0f*ui*isum + ui*ui*WINSZ, EPSV);
        float jvar  = fmaxf(j2s - 2.0f*uj*jsum + uj*uj*WINSZ, EPSV);
        local += (cross * cross) / (ivar * jvar);
    }
    if (tid == 0) redsum = 0.0f;
    __syncthreads();
    atomicAdd(&redsum, local);
    __syncthreads();
    if (tid == 0) {
        const float inv_total = 1.0f / (float)((long)NB*DD*HH*WW);
        atomicAdd(out, -redsum * inv_total);
    }
}

extern "C" void kernel_launch(void* const* d_in, const int* in_sizes, int n_in,
                              void* d_out, int out_size, void* d_ws, size_t ws_size,
                              hipStream_t stream)
{
    const float* I = (const float*)d_in[0];
    const float* J = (const float*)d_in[1];
    float* out = (float*)d_out;
    (void)hipMemsetAsync(out, 0, sizeof(float), stream);
    hipLaunchKernelGGL(ncc_fused_kernel, dim3(NBLK), dim3(NT), 0, stream, I, J, out);
}